// Conv2d_72146860638242
// MI455X (gfx1250) — compile-verified
//
#include <hip/hip_runtime.h>

typedef float v2f __attribute__((ext_vector_type(2)));
typedef float v8f __attribute__((ext_vector_type(8)));

#define K_TAPS 49
#define K_PAD  52          // 13 x 4, zero padded
#define C_OUT  32
#define WAVES_PER_BLOCK 8  // wave32: 256 threads = 8 waves
#define BLOCK_THREADS   256
#define WIN_PER_WAVE    16
#define WIN_PER_BLOCK   (WAVES_PER_BLOCK * WIN_PER_WAVE)

// One wave computes a [32 channels x 16 windows] fp32 output tile via
// V_WMMA_F32_16X16X4_F32: A = weight tile (M=ch, K=4 taps), B = enc_x^T tile
// (K=4 taps, N=16 windows), two channel-halves share one B operand.
__global__ __launch_bounds__(BLOCK_THREADS)
void conv2d_im2col_wmma_f32(const float* __restrict__ enc_x,
                            const float* __restrict__ weight,
                            const float* __restrict__ bias,
                            float* __restrict__ out,
                            int W)
{
    __shared__ float s_w[C_OUT * K_PAD];                      // weights, K zero-padded to 52
    __shared__ float s_b[C_OUT];                              // bias
    __shared__ float s_x[WAVES_PER_BLOCK][WIN_PER_WAVE * K_PAD]; // per-wave enc_x tile

    const int tid  = threadIdx.x;
    const int wave = tid >> 5;       // wave32
    const int lane = tid & 31;
    const int m    = lane & 15;      // row-within-tile (A: channel%16, B: window%16)
    const int koff = (lane & 16) ? 2 : 0;  // ISA A/B layout: upper half-wave holds K=2,3

    // Stage weights into LDS, padded with zeros for taps 49..51.
    for (int i = tid; i < C_OUT * K_PAD; i += BLOCK_THREADS) {
        int r = i / K_PAD;
        int c = i - r * K_PAD;
        s_w[i] = (c < K_TAPS) ? weight[r * K_TAPS + c] : 0.0f;
    }
    if (tid < C_OUT) s_b[tid] = bias[tid];

    const int wtile  = (blockIdx.x * WAVES_PER_BLOCK + wave) * WIN_PER_WAVE;
    const bool valid = (wtile + WIN_PER_WAVE) <= W;   // wave-uniform

    if (valid) {
        // 16 rows x 49 taps is one contiguous 3136B span -> coalesced staging.
        const float* src = enc_x + (size_t)wtile * K_TAPS;
        for (int i = lane; i < WIN_PER_WAVE * K_TAPS; i += 32) {
            int r = i / K_TAPS;
            int c = i - r * K_TAPS;
            s_x[wave][r * K_PAD + c] = src[i];
        }
        if (lane < WIN_PER_WAVE) {   // zero the K padding
            s_x[wave][lane * K_PAD + 49] = 0.0f;
            s_x[wave][lane * K_PAD + 50] = 0.0f;
            s_x[wave][lane * K_PAD + 51] = 0.0f;
        }
    }
    __syncthreads();
    if (!valid) return;              // wave-uniform: EXEC stays all-ones for WMMA

    v8f c0 = {};  // channels 0..15  x windows
    v8f c1 = {};  // channels 16..31 x windows

    const float* xrow  = &s_x[wave][m * K_PAD + koff];        // B: enc_x[window m]
    const float* a0row = &s_w[(m)      * K_PAD + koff];       // A: weight[ch m]
    const float* a1row = &s_w[(m + 16) * K_PAD + koff];       // A: weight[ch m+16]

#pragma unroll
    for (int kb = 0; kb < K_PAD; kb += 4) {                   // 13 K-steps
        v2f b  = *(const v2f*)(xrow  + kb);                   // ds_load_b64 (8B aligned)
        v2f a0 = *(const v2f*)(a0row + kb);
        v2f a1 = *(const v2f*)(a1row + kb);
        c0 = __builtin_amdgcn_wmma_f32_16x16x4_f32(
                 false, a0, false, b, (short)0, c0, false, false);
        c1 = __builtin_amdgcn_wmma_f32_16x16x4_f32(
                 false, a1, false, b, (short)0, c1, false, false);
    }

    // D layout (ISA 7.12.2): VGPR r, lane L -> channel = r + (L>=16 ? 8 : 0),
    // window = L & 15. Add bias and store channel-major [C * W].
    const int halfsel = (lane & 16) ? 8 : 0;
    float* o = out + (size_t)wtile + m;
#pragma unroll
    for (int r = 0; r < 8; ++r) {
        int ch0 = r + halfsel;
        int ch1 = ch0 + 16;
        o[(size_t)ch0 * W] = c0[r] + s_b[ch0];
        o[(size_t)ch1 * W] = c1[r] + s_b[ch1];
    }
}

extern "C" void kernel_launch(void* const* d_in, const int* in_sizes, int n_in,
                              void* d_out, int out_size, void* d_ws, size_t ws_size,
                              hipStream_t stream) {
    const float* enc_x  = (const float*)d_in[0];   // [W, 49]
    const float* weight = (const float*)d_in[1];   // [32, 7, 7] flat
    const float* bias   = (const float*)d_in[2];   // [32]
    float* out = (float*)d_out;                    // [32 * W]

    const int W = in_sizes[0] / K_TAPS;
    const int blocks = (W + WIN_PER_BLOCK - 1) / WIN_PER_BLOCK;

    conv2d_im2col_wmma_f32<<<blocks, BLOCK_THREADS, 0, stream>>>(
        enc_x, weight, bias, out, W);
}